// LSTMRecurrentModel_429496729926
// MI455X (gfx1250) — compile-verified
//
#include <hip/hip_runtime.h>
#include <math.h>

// Problem constants (match reference)
#define DD     512
#define FOURD  2048
#define LSEQ   32768
#define NWG    16                       // one CDNA5 cluster's worth of workgroups
#define CPW    (FOURD / NWG)            // 128 gate columns per workgroup

typedef __attribute__((ext_vector_type(2))) float    v2f;
typedef __attribute__((ext_vector_type(8))) float    v8f;
typedef __attribute__((ext_vector_type(4))) unsigned u32x4;
typedef __attribute__((ext_vector_type(8))) int      i32x8;
typedef __attribute__((ext_vector_type(4))) int      i32x4;

// ---------------------------------------------------------------------------
// Kernel A: x_proj = xs @ Wi + b  via V_WMMA_F32_16X16X4_F32.
// Block = 8 waves covering 8 M-tiles of one N-tile; the shared 512x16 Wi
// panel is staged in LDS once per block. HBM-bound (~324 MB), fp32 keeps
// reference numerics for free.
// ---------------------------------------------------------------------------
__global__ __launch_bounds__(256) void lstm_xproj_wmma(
    const float* __restrict__ xs, const float* __restrict__ Wi,
    const float* __restrict__ bias, float* __restrict__ xproj)
{
  __shared__ float btile[DD * 16];                 // Wi[:, n0:n0+16], 32 KB

  const int tid  = threadIdx.x;
  const int lane = tid & 31;
  const int wave = tid >> 5;
  const int mblk  = blockIdx.x >> 7;               // 256 row-blocks of 128
  const int ntile = blockIdx.x & 127;              // 128 N-tiles
  const int n0 = ntile * 16;
  const int m0 = mblk * 128 + wave * 16;

  // stage B panel (coalesced within 16-float rows)
  for (int idx = tid; idx < DD * 16; idx += 256) {
    int k = idx >> 4, c = idx & 15;
    btile[idx] = Wi[(size_t)k * FOURD + n0 + c];
  }
  __syncthreads();

  // A-frag (16x4 f32): lanes 0-15 hold M=lane,K={0,1}; lanes 16-31 K={2,3}
  const int row = lane & 15;
  const int kb  = (lane >> 4) << 1;                // 0 or 2
  const int col = lane & 15;

  const float* arow = xs + (size_t)(m0 + row) * DD + kb;

  v8f acc = {};
  for (int k0 = 0; k0 < DD; k0 += 4) {
    v2f a = *(const v2f*)(arow + k0);              // 8B-aligned pair
    v2f b;
    b.x = btile[(k0 + kb) * 16 + col];             // conflict-free ds_load_b32
    b.y = btile[(k0 + kb + 1) * 16 + col];
    // 8 args: (neg_a, A, neg_b, B, c_mod, C, reuse_a, reuse_b)
    acc = __builtin_amdgcn_wmma_f32_16x16x4_f32(false, a, false, b,
                                                (short)0, acc, false, false);
  }

  // C/D layout: lanes 0-15 -> M=m0+r, lanes 16-31 -> M=m0+8+r; N = n0+(lane&15)
  const float bv = bias[n0 + col];
  const int orow = m0 + ((lane >> 4) << 3);
  float* outp = xproj + (size_t)orow * FOURD + (n0 + col);
#pragma unroll
  for (int r = 0; r < 8; ++r)
    outp[(size_t)r * FOURD] = acc[r] + bv;
}

// ---------------------------------------------------------------------------
// Init kernel: reset inter-workgroup barrier state each call (d_ws is poisoned).
// ---------------------------------------------------------------------------
__global__ void lstm_init(unsigned* bar)
{
  if (threadIdx.x == 0) { bar[0] = 0u; bar[1] = 0u; }
}

// ---------------------------------------------------------------------------
// Kernel B: persistent recurrent scan. 16 workgroups; each DMAs its 512x128
// slice of Wh into LDS (256 KB of the 320 KB WGP LDS) with the Tensor Data
// Mover, keeps (h,c) in LDS, and per step exchanges the 2048 gate
// pre-activations through an L2-resident double buffer with agent-scope
// atomics + a device barrier (cluster split-barrier when clustered, NOP
// otherwise, plus a correct atomic sense-reversal barrier).
// ---------------------------------------------------------------------------
__global__ __launch_bounds__(256) void lstm_recurrent(
    const float* __restrict__ Wh, const float* __restrict__ c0,
    const float* __restrict__ h0, const float* __restrict__ xproj,
    float* __restrict__ ybuf,      // 2 * FOURD floats (double buffer)
    unsigned* __restrict__ bar,    // [count, generation]
    float* __restrict__ out)       // [LSEQ, DD]
{
  extern __shared__ float smem[];
  float* wh_l = smem;                  // [512][128] k-major, 65536 floats
  float* h_l  = wh_l + DD * CPW;       // DD
  float* c_l  = h_l + DD;              // DD
  float* part = c_l + DD;              // 8 * 128 partials

  const int tid = threadIdx.x;
  const int wg  = blockIdx.x;
  const int colbase = wg * CPW;

  // ---- TDM preload: 2D tile (tile_dim0=128 cols, tile_dim1=512 rows),
  // row stride 2048 elements, 4B data, into contiguous k-major LDS.
  {
    unsigned long long ga = (unsigned long long)(const void*)(Wh + colbase);
    u32x4 g0;
    g0[0] = 1u;                                          // count=1 valid D#
    g0[1] = (unsigned)(unsigned long long)(void*)wh_l;   // lds_addr (bytes)
    g0[2] = (unsigned)(ga & 0xFFFFFFFFull);              // global_addr[31:0]
    g0[3] = (unsigned)((ga >> 32) & 0x1FFFFFFull)        // global_addr[56:32]
          | (2u << 30);                                  // type = 2 ("image")
    i32x8 g1;
    g1[0] = (int)(2u << 16);          // data_size = 4B; no mask/pad/iterate
    g1[1] = (int)((unsigned)FOURD << 16);   // tensor_dim0 = 2048 (lo16)
    g1[2] = (int)((unsigned)DD << 16);      // dim0 hi=0 | tensor_dim1 = 512
    g1[3] = (int)((unsigned)CPW << 16);     // dim1 hi=0 | tile_dim0 = 128
    g1[4] = (int)DD;                        // tile_dim1 = 512, tile_dim2 = 0
    g1[5] = (int)FOURD;                     // tensor_dim0_stride = 2048
    g1[6] = 0;
    g1[7] = 0;
    i32x4 gz4 = {0, 0, 0, 0};
    i32x8 gz8 = {0, 0, 0, 0, 0, 0, 0, 0};
    if (tid == 0)                            // one wave issues the DMA
      __builtin_amdgcn_tensor_load_to_lds(g0, g1, gz4, gz4, gz8, 0);
  }
  for (int j = tid; j < DD; j += 256) { h_l[j] = h0[j]; c_l[j] = c0[j]; }
  __builtin_amdgcn_s_wait_tensorcnt((short)0);   // wave 0 drains its TDM op
  __syncthreads();                               // publishes LDS to the block

  // Each thread: 4 adjacent columns (cgrp) x 64 k-values (ks slice).
  const int cgrp  = tid & 31;
  const int ks    = tid >> 5;                    // wave index == k-slice
  const int kbase = ks * 64;

  unsigned gen = 0;
  for (int t = 0; t < LSEQ; ++t) {
    // ---- phase 1: partial matvec. Lanes read 4 consecutive columns of a
    // Wh row (ds_load_b128, all 64 banks hit exactly twice); h[k] broadcasts.
    float4 a4 = {0.f, 0.f, 0.f, 0.f};
    const float4* hp = (const float4*)(h_l + kbase);
#pragma unroll 4
    for (int kk = 0; kk < 16; ++kk) {
      float4 hv = hp[kk];
      const float* wrow = wh_l + (kbase + kk * 4) * CPW + (cgrp << 2);
      float4 w0 = *(const float4*)(wrow);
      float4 w1 = *(const float4*)(wrow + CPW);
      float4 w2 = *(const float4*)(wrow + 2 * CPW);
      float4 w3 = *(const float4*)(wrow + 3 * CPW);
      a4.x += hv.x * w0.x + hv.y * w1.x + hv.z * w2.x + hv.w * w3.x;
      a4.y += hv.x * w0.y + hv.y * w1.y + hv.z * w2.y + hv.w * w3.y;
      a4.z += hv.x * w0.z + hv.y * w1.z + hv.z * w2.z + hv.w * w3.z;
      a4.w += hv.x * w0.w + hv.y * w1.w + hv.z * w2.w + hv.w * w3.w;
    }
    *(float4*)(part + ks * CPW + (cgrp << 2)) = a4;
    __syncthreads();

    float* ywr = ybuf + (size_t)(t & 1) * FOURD;
    if (tid < CPW) {
      float y = xproj[(size_t)t * FOURD + colbase + tid];
#pragma unroll
      for (int s = 0; s < 8; ++s) y += part[s * CPW + tid];
      __hip_atomic_store(&ywr[colbase + tid], y,
                         __ATOMIC_RELEASE, __HIP_MEMORY_SCOPE_AGENT);
      if (t + 1 < LSEQ)   // pull next step's activations toward the WGP
        __builtin_prefetch(&xproj[(size_t)(t + 1) * FOURD + colbase + tid], 0, 1);
    }
    __threadfence();      // each storing thread drains its store to device scope
    __syncthreads();

    // ---- inter-workgroup barrier (once per step)
    // Cluster split-barrier: NOP when not launched as a cluster (ISA 08 §2).
    __builtin_amdgcn_s_cluster_barrier();
    if (tid == 0) {
      unsigned arr = __hip_atomic_fetch_add(&bar[0], 1u,
                        __ATOMIC_ACQ_REL, __HIP_MEMORY_SCOPE_AGENT);
      if (arr == NWG - 1) {
        __hip_atomic_store(&bar[0], 0u, __ATOMIC_RELAXED, __HIP_MEMORY_SCOPE_AGENT);
        __hip_atomic_store(&bar[1], gen + 1, __ATOMIC_RELEASE, __HIP_MEMORY_SCOPE_AGENT);
      } else {
        while ((int)(__hip_atomic_load(&bar[1], __ATOMIC_ACQUIRE,
                                       __HIP_MEMORY_SCOPE_AGENT) - (gen + 1)) < 0)
          __builtin_amdgcn_s_sleep(2);
      }
    }
    gen++;
    __syncthreads();

    // ---- phase 2: every WG redundantly updates its full (c,h) copy
    const float* yrd = ybuf + (size_t)(t & 1) * FOURD;
#pragma unroll
    for (int q = 0; q < 2; ++q) {
      int j = tid + q * 256;
      float iv = __hip_atomic_load(&yrd[j],          __ATOMIC_RELAXED, __HIP_MEMORY_SCOPE_AGENT);
      float fv = __hip_atomic_load(&yrd[DD + j],     __ATOMIC_RELAXED, __HIP_MEMORY_SCOPE_AGENT);
      float gv = __hip_atomic_load(&yrd[2 * DD + j], __ATOMIC_RELAXED, __HIP_MEMORY_SCOPE_AGENT);
      float ov = __hip_atomic_load(&yrd[3 * DD + j], __ATOMIC_RELAXED, __HIP_MEMORY_SCOPE_AGENT);
      float si = 1.f / (1.f + __expf(-iv));
      float sf = 1.f / (1.f + __expf(-fv));
      float so = 1.f / (1.f + __expf(-ov));
      float cn = sf * c_l[j] + si * tanhf(gv);
      float hn = so * tanhf(cn);
      c_l[j] = cn;
      h_l[j] = hn;
      if (wg == 0) out[(size_t)t * DD + j] = hn;   // all copies identical
    }
    __syncthreads();   // h_l ready for next step's matvec
  }
}

// ---------------------------------------------------------------------------
extern "C" void kernel_launch(void* const* d_in, const int* in_sizes, int n_in,
                              void* d_out, int out_size, void* d_ws, size_t ws_size,
                              hipStream_t stream)
{
  (void)in_sizes; (void)n_in; (void)out_size; (void)ws_size;
  const float* xs = (const float*)d_in[0];
  const float* Wi = (const float*)d_in[1];
  const float* Wh = (const float*)d_in[2];
  const float* b  = (const float*)d_in[3];
  const float* c0 = (const float*)d_in[4];
  const float* h0 = (const float*)d_in[5];
  float* out = (float*)d_out;

  char* ws = (char*)d_ws;
  float*    xproj = (float*)ws;                                        // L*4D f32 = 256 MB
  float*    ybuf  = (float*)(ws + (size_t)LSEQ * FOURD * sizeof(float));
  unsigned* bar   = (unsigned*)(ws + (size_t)LSEQ * FOURD * sizeof(float)
                                   + (size_t)2 * FOURD * sizeof(float));

  lstm_init<<<1, 64, 0, stream>>>(bar);

  // 256 M-blocks (of 128 rows) x 128 N-tiles
  lstm_xproj_wmma<<<256 * 128, 256, 0, stream>>>(xs, Wi, b, xproj);

  const size_t smem = (size_t)(DD * CPW + DD + DD + 8 * CPW) * sizeof(float);
  lstm_recurrent<<<NWG, 256, smem, stream>>>(Wh, c0, h0, xproj, ybuf, bar, out);
}